// YOLOLoss_63677185130639
// MI455X (gfx1250) — compile-verified
//
#include <hip/hip_runtime.h>
#include <cstdint>

#define LAMBDA_COORD 5.0f
#define LAMBDA_NOOBJ 0.5f
#define CELLS_PER_BLOCK 256
#define FLOATS_PER_CELL 30
#define BLOCK_FLOATS (CELLS_PER_BLOCK * FLOATS_PER_CELL)   // 7680
#define BLOCK_F4     (BLOCK_FLOATS / 4)                    // 1920

typedef float v2f __attribute__((ext_vector_type(2)));
typedef float v8f __attribute__((ext_vector_type(8)));

// Wave32 sum via one V_WMMA_F32_16X16X4_F32:
//   A[m][0]=p_m (lanes 0-15, vgpr0), A[m][2]=p_{m+16} (lanes 16-31, vgpr0), A[*][1]=A[*][3]=0
//   B = ones(4x16)  =>  D[m][n] = p_m + p_{m+16}  for all n
//   lane n holds D rows 0..7 (n<16) or 8..15 (n>=16) in c[0..7];
//   sum(c) + shfl_xor(.,16) = full 32-lane sum, uniform across the wave.
__device__ __forceinline__ float wave_sum_wmma(float p) {
  v2f a; a[0] = p;    a[1] = 0.0f;
  v2f b; b[0] = 1.0f; b[1] = 1.0f;
  v8f c = {};
  c = __builtin_amdgcn_wmma_f32_16x16x4_f32(false, a, false, b, (short)0, c, false, false);
  float s = c[0] + c[1] + c[2] + c[3] + c[4] + c[5] + c[6] + c[7];
  s += __shfl_xor(s, 16, 32);
  return s;
}

__device__ __forceinline__ float iou_xywh(const float* b1, const float* b2) {
  float b1x1 = b1[0] - b1[2] * 0.5f, b1y1 = b1[1] - b1[3] * 0.5f;
  float b1x2 = b1[0] + b1[2] * 0.5f, b1y2 = b1[1] + b1[3] * 0.5f;
  float b2x1 = b2[0] - b2[2] * 0.5f, b2y1 = b2[1] - b2[3] * 0.5f;
  float b2x2 = b2[0] + b2[2] * 0.5f, b2y2 = b2[1] + b2[3] * 0.5f;
  float iw = fmaxf(fminf(b1x2, b2x2) - fmaxf(b1x1, b2x1), 0.0f);
  float ih = fmaxf(fminf(b1y2, b2y2) - fmaxf(b1y1, b2y1), 0.0f);
  float inter = iw * ih;
  float a1 = (b1x2 - b1x1) * (b1y2 - b1y1);
  float a2 = (b2x2 - b2x1) * (b2y2 - b2y1);
  return inter / (a1 + a2 - inter + 1e-10f);
}

__global__ void __launch_bounds__(256)
yolo_partial_kernel(const float* __restrict__ preds,
                    const float* __restrict__ labels,
                    float* __restrict__ partial,
                    int numCells, long long totalFloats) {
  __shared__ float lds[2 * BLOCK_FLOATS];   // [0..7679]=preds chunk, [7680..]=labels chunk
  __shared__ float wsum[8];

  const int tid = threadIdx.x;
  const long long blockFloatBase = (long long)blockIdx.x * BLOCK_FLOATS;

  // LDS byte offsets (addrspace(3) offset lives in the low 32 bits of the flat ptr)
  unsigned ldsP = (unsigned)(uintptr_t)(&lds[0]);
  unsigned ldsL = (unsigned)(uintptr_t)(&lds[BLOCK_FLOATS]);
  unsigned long long gP = (unsigned long long)(uintptr_t)preds  + (unsigned long long)blockFloatBase * 4ull;
  unsigned long long gL = (unsigned long long)(uintptr_t)labels + (unsigned long long)blockFloatBase * 4ull;

  // Async bulk copy global -> LDS, 128b per lane, fully coalesced (block base is 16B aligned).
  for (int i = tid; i < BLOCK_F4; i += 256) {
    if (blockFloatBase + (long long)i * 4 + 4 <= totalFloats) {
      unsigned off = (unsigned)i * 16u;
      asm volatile("global_load_async_to_lds_b128 %0, %1, %2"
                   :: "v"(ldsP + off), "v"(off), "s"(gP) : "memory");
      asm volatile("global_load_async_to_lds_b128 %0, %1, %2"
                   :: "v"(ldsL + off), "v"(off), "s"(gL) : "memory");
    }
  }
  asm volatile("s_wait_asynccnt 0" ::: "memory");
  __syncthreads();

  float contrib = 0.0f;
  const int cell = blockIdx.x * CELLS_PER_BLOCK + tid;
  if (cell < numCells) {
    const float* p = &lds[tid * FLOATS_PER_CELL];
    const float* l = &lds[BLOCK_FLOATS + tid * FLOATS_PER_CELL];

    const bool obj = (l[4] == 1.0f);
    const float iou1 = iou_xywh(p + 0, l + 0);
    const float iou2 = iou_xywh(p + 5, l + 0);
    const bool box1 = iou1 > iou2;

    float dx1 = l[0] - p[0], dy1 = l[1] - p[1];
    float xy1 = dx1 * dx1 + dy1 * dy1;
    float dx2 = l[5] - p[5], dy2 = l[6] - p[6];
    float xy2 = dx2 * dx2 + dy2 * dy2;

    float sw1 = sqrtf(l[2]) - sqrtf(p[2]), sh1 = sqrtf(l[3]) - sqrtf(p[3]);
    float wh1 = sw1 * sw1 + sh1 * sh1;
    float sw2 = sqrtf(l[7]) - sqrtf(p[7]), sh2 = sqrtf(l[8]) - sqrtf(p[8]);
    float wh2 = sw2 * sw2 + sh2 * sh2;

    float c1 = l[4] - p[4]; c1 *= c1;
    float c2 = l[9] - p[9]; c2 *= c2;
    const float p4 = p[4], p9 = p[9];

    float cls = 0.0f;
    #pragma unroll
    for (int k = 10; k < 30; ++k) { float d = l[k] - p[k]; cls += d * d; }

    const float sel_xy = box1 ? xy1 : xy2;
    const float sel_wh = box1 ? wh1 : wh2;
    const float sel_cf = box1 ? c1 : c2;
    const float noobj  = obj ? (box1 ? p9 * p9 : p4 * p4) : (p4 * p4 + p9 * p9);
    const float o = obj ? 1.0f : 0.0f;

    contrib = o * (LAMBDA_COORD * (sel_xy + sel_wh) + sel_cf + cls) + LAMBDA_NOOBJ * noobj;
  }

  // EXEC is all-ones here (reconverged) -> WMMA wave reduction is legal.
  const float ws = wave_sum_wmma(contrib);
  if ((tid & 31) == 0) wsum[tid >> 5] = ws;
  __syncthreads();
  if (tid == 0) {
    float t = 0.0f;
    #pragma unroll
    for (int i = 0; i < 8; ++i) t += wsum[i];
    partial[blockIdx.x] = t;
  }
}

__global__ void __launch_bounds__(256)
yolo_reduce_kernel(const float* __restrict__ partial, int n,
                   float* __restrict__ out, float invB) {
  __shared__ float wsum[8];
  const int tid = threadIdx.x;
  float s = 0.0f;
  for (int i = tid; i < n; i += 256) s += partial[i];   // fixed order -> deterministic
  const float w = wave_sum_wmma(s);
  if ((tid & 31) == 0) wsum[tid >> 5] = w;
  __syncthreads();
  if (tid == 0) {
    float t = 0.0f;
    #pragma unroll
    for (int i = 0; i < 8; ++i) t += wsum[i];
    out[0] = t * invB;
  }
}

extern "C" void kernel_launch(void* const* d_in, const int* in_sizes, int n_in,
                              void* d_out, int out_size, void* d_ws, size_t ws_size,
                              hipStream_t stream) {
  const float* preds  = (const float*)d_in[0];
  const float* labels = (const float*)d_in[1];
  const long long totalFloats = (long long)in_sizes[0];       // B*7*7*30
  const int numCells = (int)(totalFloats / FLOATS_PER_CELL);  // B*49
  const int B = numCells / 49;
  const int nblocks = (numCells + CELLS_PER_BLOCK - 1) / CELLS_PER_BLOCK;

  float* partial = (float*)d_ws;
  yolo_partial_kernel<<<nblocks, 256, 0, stream>>>(preds, labels, partial, numCells, totalFloats);
  yolo_reduce_kernel<<<1, 256, 0, stream>>>(partial, nblocks, (float*)d_out, 1.0f / (float)B);
}